// OutputBlock_31842887533378
// MI455X (gfx1250) — compile-verified
//
#include <hip/hip_runtime.h>
#include <math.h>

#define EMB 128
#define NUM_RADIAL 6
#define NUM_TARGETS 12
#define LDS_STRIDE 132

typedef __attribute__((ext_vector_type(2))) float v2f;
typedef __attribute__((ext_vector_type(4))) float v4f;
typedef __attribute__((ext_vector_type(8))) float v8f;

// ---------------------------------------------------------------------------
// Native no-return f32 global atomic add at device scope (L2 RMW).
// ---------------------------------------------------------------------------
__device__ __forceinline__ void atomic_add_f32_dev(float* p, float v) {
  unsigned long long a = (unsigned long long)p;
  asm volatile("global_atomic_add_f32 %0, %1, off scope:SCOPE_DEV"
               :: "v"(a), "v"(v) : "memory");
}

// ---------------------------------------------------------------------------
// WMMA f32 16x16x4 wrapper (8-arg VOP3P form).
// ---------------------------------------------------------------------------
__device__ __forceinline__ v8f wmma_f32x4(v2f a, v2f b, v8f c) {
  return __builtin_amdgcn_wmma_f32_16x16x4_f32(
      /*neg_a=*/false, a, /*neg_b=*/false, b,
      /*c_mod=*/(short)0, c, /*reuse_a=*/false, /*reuse_b=*/false);
}

// ---------------------------------------------------------------------------
// Kernel 1: zero the atom accumulator (float4 stores).
// ---------------------------------------------------------------------------
__global__ void zero_kernel(v4f* __restrict__ h4, int n4) {
  int i = blockIdx.x * blockDim.x + threadIdx.x;
  if (i < n4) h4[i] = v4f{0.f, 0.f, 0.f, 0.f};
}

// ---------------------------------------------------------------------------
// Kernel 2: edge gate + scatter-add.
// One wave per edge: lane handles 4 consecutive emb columns (float4 loads).
// Streaming inputs (x, rbf, idnb) use non-temporal loads so the 51 MB atom
// accumulator stays resident in the 192 MB L2 while 865 MB streams past.
// ---------------------------------------------------------------------------
__global__ __launch_bounds__(256)
void edge_gate_scatter(const float* __restrict__ x,
                       const float* __restrict__ rbf,
                       const int*   __restrict__ idnb,
                       const float* __restrict__ Wrbf,
                       float*       __restrict__ h,
                       int n_edges) {
  int wave = (blockIdx.x * blockDim.x + threadIdx.x) >> 5;
  int lane = threadIdx.x & 31;
  if (wave >= n_edges) return;
  const int e = wave;

  // wave-uniform loads (streaming -> NT)
  int atom = __builtin_nontemporal_load(&idnb[e]);
  float r0 = __builtin_nontemporal_load(&rbf[e * NUM_RADIAL + 0]);
  float r1 = __builtin_nontemporal_load(&rbf[e * NUM_RADIAL + 1]);
  float r2 = __builtin_nontemporal_load(&rbf[e * NUM_RADIAL + 2]);
  float r3 = __builtin_nontemporal_load(&rbf[e * NUM_RADIAL + 3]);
  float r4 = __builtin_nontemporal_load(&rbf[e * NUM_RADIAL + 4]);
  float r5 = __builtin_nontemporal_load(&rbf[e * NUM_RADIAL + 5]);

  // W_rbf is [6][128]; lane owns cols lane*4 .. lane*4+3 (hot, keep cached)
  const v4f* W4 = (const v4f*)Wrbf;   // [6][32] float4
  v4f w0 = W4[0 * 32 + lane];
  v4f w1 = W4[1 * 32 + lane];
  v4f w2 = W4[2 * 32 + lane];
  v4f w3 = W4[3 * 32 + lane];
  v4f w4 = W4[4 * 32 + lane];
  v4f w5 = W4[5 * 32 + lane];

  v4f g = r0 * w0 + r1 * w1 + r2 * w2 + r3 * w3 + r4 * w4 + r5 * w5;

  v4f xv = __builtin_nontemporal_load(((const v4f*)x) + e * (EMB / 4) + lane);
  v4f m = g * xv;

  float* dst = h + (size_t)atom * EMB + lane * 4;
  atomic_add_f32_dev(dst + 0, m.x);
  atomic_add_f32_dev(dst + 1, m.y);
  atomic_add_f32_dev(dst + 2, m.z);
  atomic_add_f32_dev(dst + 3, m.w);
}

// ---------------------------------------------------------------------------
// One dense layer (128x128) + bias + swish over the 64x128 LDS tile.
// Inlined with direct kernarg pointers so W/b loads stay in the GLOBAL
// pipeline (LOADcnt), not FLAT (LOADcnt+DScnt).
// ---------------------------------------------------------------------------
__device__ __forceinline__ void mlp_layer(const float* __restrict__ W,
                                          const float* __restrict__ b,
                                          float* __restrict__ buf,
                                          int l16, int half, int n0) {
  v8f acc[4] = {v8f{}, v8f{}, v8f{}, v8f{}};

#pragma unroll 4
  for (int kt = 0; kt < EMB / 4; ++kt) {
    const int k0 = kt * 4 + half * 2;          // this lane's K pair
    v2f bfrag;
    bfrag.x = W[(k0 + 0) * EMB + n0];
    bfrag.y = W[(k0 + 1) * EMB + n0];
#pragma unroll
    for (int rt = 0; rt < 4; ++rt) {
      v2f afrag = *(const v2f*)&buf[(rt * 16 + l16) * LDS_STRIDE + k0];
      acc[rt] = wmma_f32x4(afrag, bfrag, acc[rt]);
    }
  }
  __syncthreads();   // everyone done reading buf before it is overwritten

  const float bias = b[n0];
#pragma unroll
  for (int rt = 0; rt < 4; ++rt) {
#pragma unroll
    for (int v = 0; v < 8; ++v) {
      float val = acc[rt][v] + bias;
      float sw  = val * (1.0f / (1.0f + __expf(-val)));   // swish
      buf[(rt * 16 + v + 8 * half) * LDS_STRIDE + n0] = sw;
    }
  }
  __syncthreads();
}

// ---------------------------------------------------------------------------
// Kernel 3: fused 3-layer MLP (swish) + 128x12 head, WMMA f32 16x16x4.
// Block = 256 threads (8 waves) handles 64 atom rows.
// LDS tile: 64 x 128 f32, row stride 132 (conflict-free).
// ---------------------------------------------------------------------------
__global__ __launch_bounds__(256)
void mlp_head_kernel(const float* __restrict__ h0,
                     const float* __restrict__ W1, const float* __restrict__ b1,
                     const float* __restrict__ W2, const float* __restrict__ b2,
                     const float* __restrict__ W3, const float* __restrict__ b3,
                     const float* __restrict__ Wf,
                     float* __restrict__ out,
                     int n_atoms) {
  __shared__ float buf[64 * LDS_STRIDE];

  const int tid  = threadIdx.x;
  const int wave = tid >> 5;
  const int lane = tid & 31;
  const int half = lane >> 4;      // 0: lanes 0-15, 1: lanes 16-31
  const int l16  = lane & 15;
  const int rowBase = blockIdx.x * 64;

  // stage 64x128 activation tile into LDS (zero-pad tail rows)
  for (int i = tid; i < 64 * 32; i += 256) {
    int r  = i >> 5;
    int c4 = i & 31;
    int atom = rowBase + r;
    v4f v = v4f{0.f, 0.f, 0.f, 0.f};
    if (atom < n_atoms) v = ((const v4f*)h0)[(size_t)atom * 32 + c4];
    *(v4f*)&buf[r * LDS_STRIDE + c4 * 4] = v;
  }
  __syncthreads();

  const int n0 = wave * 16 + l16;  // output column owned by this lane

  mlp_layer(W1, b1, buf, l16, half, n0);
  mlp_layer(W2, b2, buf, l16, half, n0);
  mlp_layer(W3, b3, buf, l16, half, n0);

  // final head: 128 -> 12, B tile zero-padded to 16 cols; waves 0..3 (uniform)
  if (wave < 4) {
    const int rt = wave;
    v8f acc = v8f{};
#pragma unroll 4
    for (int kt = 0; kt < EMB / 4; ++kt) {
      const int k0 = kt * 4 + half * 2;
      v2f bfrag;
      bfrag.x = (l16 < NUM_TARGETS) ? Wf[(k0 + 0) * NUM_TARGETS + l16] : 0.f;
      bfrag.y = (l16 < NUM_TARGETS) ? Wf[(k0 + 1) * NUM_TARGETS + l16] : 0.f;
      v2f afrag = *(const v2f*)&buf[(rt * 16 + l16) * LDS_STRIDE + k0];
      acc = wmma_f32x4(afrag, bfrag, acc);
    }
#pragma unroll
    for (int v = 0; v < 8; ++v) {
      int row  = rt * 16 + v + 8 * half;
      int atom = rowBase + row;
      if (atom < n_atoms && l16 < NUM_TARGETS)
        __builtin_nontemporal_store(acc[v], &out[(size_t)atom * NUM_TARGETS + l16]);
    }
  }
}

// ---------------------------------------------------------------------------
// launcher
// ---------------------------------------------------------------------------
extern "C" void kernel_launch(void* const* d_in, const int* in_sizes, int n_in,
                              void* d_out, int out_size, void* d_ws, size_t ws_size,
                              hipStream_t stream) {
  const float* x     = (const float*)d_in[0];
  const float* rbf   = (const float*)d_in[1];
  const int*   idnb  = (const int*)  d_in[2];
  // d_in[3] = n_atoms scalar (device); sizes derived on host instead
  const float* W_rbf = (const float*)d_in[4];
  const float* W1    = (const float*)d_in[5];
  const float* b1    = (const float*)d_in[6];
  const float* W2    = (const float*)d_in[7];
  const float* b2    = (const float*)d_in[8];
  const float* W3    = (const float*)d_in[9];
  const float* b3    = (const float*)d_in[10];
  const float* Wf    = (const float*)d_in[11];
  float* out = (float*)d_out;

  const int n_edges = in_sizes[0] / EMB;
  const int n_atoms = out_size / NUM_TARGETS;

  float* h = (float*)d_ws;   // n_atoms * EMB floats (51.2 MB) — L2-resident

  // 1) zero accumulator
  {
    int n4 = (n_atoms * EMB) / 4;
    int blocks = (n4 + 255) / 256;
    zero_kernel<<<blocks, 256, 0, stream>>>((v4f*)h, n4);
  }
  // 2) edge gate + scatter (one wave per edge)
  {
    long long threads = (long long)n_edges * 32;
    int blocks = (int)((threads + 255) / 256);
    edge_gate_scatter<<<blocks, 256, 0, stream>>>(x, rbf, idnb, W_rbf, h, n_edges);
  }
  // 3) fused MLP + head (64 rows per block)
  {
    int blocks = (n_atoms + 63) / 64;
    mlp_head_kernel<<<blocks, 256, 0, stream>>>(h, W1, b1, W2, b2, W3, b3, Wf,
                                                out, n_atoms);
  }
}